// ResNet_18038862643505
// MI455X (gfx1250) — compile-verified
//
#include <hip/hip_runtime.h>
#include <hip/hip_bf16.h>

// CDNA5 / gfx1250: wave32, fp32 WMMA 16x16x4.
typedef float v2f __attribute__((ext_vector_type(2)));
typedef float v8f __attribute__((ext_vector_type(8)));

#define HID 256
#define NCH 16            // HID / 16 hidden chunks
#define BATCH_PER_WAVE 16
#define WAVES_PER_BLOCK 8
#define THREADS (WAVES_PER_BLOCK * 32)
#define BATCH_PER_BLOCK (WAVES_PER_BLOCK * BATCH_PER_WAVE)

__device__ __forceinline__ float fast_tanh(float x) {
#if defined(__has_builtin)
#if __has_builtin(__builtin_amdgcn_tanhf)
    return __builtin_amdgcn_tanhf(x);          // v_tanh_f32 (TRANS pipe)
#elif __has_builtin(__builtin_amdgcn_tanh_f32)
    return __builtin_amdgcn_tanh_f32(x);
#else
    return tanhf(x);
#endif
#else
    return tanhf(x);
#endif
}

__global__ __launch_bounds__(THREADS) void node_rk4_wmma_kernel(
    const float* __restrict__ x,   // [B,2]
    const float* __restrict__ W1,  // [2,HID]
    const float* __restrict__ b1,  // [HID]
    const float* __restrict__ W2,  // [HID,2]
    const float* __restrict__ b2,  // [2]
    const float* __restrict__ Wf,  // [2,2]
    const float* __restrict__ bf,  // [2]
    const int*   __restrict__ tptr,
    float* __restrict__ out,       // [2*B logits | 2*B probas]
    int B)
{
    // W2 (2 KB) -> LDS once; broadcast-read every g-eval.
    __shared__ float w2s[HID * 2];
    int tid = threadIdx.x;
    w2s[tid]            = W2[tid];
    w2s[tid + THREADS]  = W2[tid + THREADS];
    __syncthreads();

    const int  lane = tid & 31;
    const int  wave = tid >> 5;
    const int  n    = lane & 15;        // batch-in-tile (C matrix N), hidden-in-chunk (A matrix M)
    const bool hiH  = lane >= 16;       // upper half-wave: carries K=2,3 rows / M=r+8 rows
    const int  m    = blockIdx.x * BATCH_PER_BLOCK + wave * BATCH_PER_WAVE + n;

    // A-matrix (W1^T chunk, 16x4, bias folded into K=2 row): loop-invariant per-lane constants.
    // Layout (ISA 32-bit A 16x4): lanes0-15 VGPR0=K0,VGPR1=K1; lanes16-31 VGPR0=K2,VGPR1=K3.
    float a0v[NCH], a1v[NCH];
#pragma unroll
    for (int c = 0; c < NCH; ++c) {
        int h = c * 16 + n;
        a0v[c] = hiH ? b1[h] : W1[h];         // K0 = W1[0][h] | K2 = b1[h]
        a1v[c] = hiH ? 0.0f : W1[HID + h];    // K1 = W1[1][h] | K3 = 0
    }
    const float b2_0 = b2[0], b2_1 = b2[1];

    // State: y for batch element (lane&15), duplicated across half-waves.
    float y0 = x[2 * m + 0];
    float y1 = x[2 * m + 1];

    // Replicate Python's `while t <= tf: n+=1; t+=h` with double accumulation.
    {
        // computed per-lane, identical everywhere; uniformize for scalar loop control
    }
    int    tv = tptr[0];
    double tfd = 0.1 * (double)tv;
    int    nst = 0;
    double tcd = 0.0;
    while (tcd <= tfd) { nst++; tcd += 0.01; }
    nst = __builtin_amdgcn_readfirstlane(nst);

    const float hstep = 0.01f;
    const float4* w2s4 = (const float4*)w2s;
    const int w2base = hiH ? 4 : 0;   // float4 offset inside a 32-dword chunk

    // g(y) = tanh(y@W1 + b1) @ W2 + b2 for the wave's 16 batch elements.
    auto geval = [&](float yy0, float yy1, float& g0, float& g1) {
        v2f bm;                       // B-matrix (4x16): K0=y0,K1=y1 (lo) | K2=1,K3=0 (hi)
        bm.x = hiH ? 1.0f : yy0;
        bm.y = hiH ? 0.0f : yy1;
        float p0 = 0.0f, p1 = 0.0f;
#pragma unroll
        for (int c = 0; c < NCH; ++c) {
            v2f am; am.x = a0v[c]; am.y = a1v[c];
            v8f acc = {0.f, 0.f, 0.f, 0.f, 0.f, 0.f, 0.f, 0.f};
            // C[16 hidden x 16 batch] = W1^T_chunk x [y;1] : v_wmma_f32_16x16x4_f32
            acc = __builtin_amdgcn_wmma_f32_16x16x4_f32(
                false, am, false, bm, (short)0, acc, false, false);
            // W2 rows for this lane's 8 hidden values h = c*16 + 8*hiH + r (broadcast per half)
            float4 wA = w2s4[c * 8 + w2base + 0];
            float4 wB = w2s4[c * 8 + w2base + 1];
            float4 wC = w2s4[c * 8 + w2base + 2];
            float4 wD = w2s4[c * 8 + w2base + 3];
            float t0 = fast_tanh(acc[0]); float t1 = fast_tanh(acc[1]);
            float t2 = fast_tanh(acc[2]); float t3 = fast_tanh(acc[3]);
            float t4 = fast_tanh(acc[4]); float t5 = fast_tanh(acc[5]);
            float t6 = fast_tanh(acc[6]); float t7 = fast_tanh(acc[7]);
            p0 = fmaf(t0, wA.x, p0); p1 = fmaf(t0, wA.y, p1);
            p0 = fmaf(t1, wA.z, p0); p1 = fmaf(t1, wA.w, p1);
            p0 = fmaf(t2, wB.x, p0); p1 = fmaf(t2, wB.y, p1);
            p0 = fmaf(t3, wB.z, p0); p1 = fmaf(t3, wB.w, p1);
            p0 = fmaf(t4, wC.x, p0); p1 = fmaf(t4, wC.y, p1);
            p0 = fmaf(t5, wC.z, p0); p1 = fmaf(t5, wC.w, p1);
            p0 = fmaf(t6, wD.x, p0); p1 = fmaf(t6, wD.y, p1);
            p0 = fmaf(t7, wD.z, p0); p1 = fmaf(t7, wD.w, p1);
        }
        // Each half summed a disjoint 128 hidden units for the SAME batch element.
        p0 += __shfl_xor(p0, 16, 32);
        p1 += __shfl_xor(p1, 16, 32);
        g0 = p0 + b2_0;
        g1 = p1 + b2_1;
    };

    for (int s = 0; s < nst; ++s) {
        float g0, g1;
        geval(y0, y1, g0, g1);
        float k10 = hstep * g0, k11 = hstep * g1;
        geval(y0 + 0.5f * k10, y1 + 0.5f * k11, g0, g1);
        float k20 = hstep * g0, k21 = hstep * g1;
        geval(y0 + 0.5f * k20, y1 + 0.5f * k21, g0, g1);
        float k30 = hstep * g0, k31 = hstep * g1;
        geval(y0 + k30, y1 + k31, g0, g1);
        float k40 = hstep * g0, k41 = hstep * g1;
        y0 += (k10 + 2.0f * k20 + 2.0f * k30 + k40) / 6.0f;
        y1 += (k11 + 2.0f * k21 + 2.0f * k31 + k41) / 6.0f;
    }

    // logits = y @ Wf + bf ; probas = softmax(logits)
    float l0 = fmaf(y1, Wf[2], fmaf(y0, Wf[0], bf[0]));
    float l1 = fmaf(y1, Wf[3], fmaf(y0, Wf[1], bf[1]));
    float mx = fmaxf(l0, l1);
    float e0 = expf(l0 - mx);
    float e1 = expf(l1 - mx);
    float inv = 1.0f / (e0 + e1);

    if (!hiH) {  // lower half writes; upper half holds identical duplicates
        out[2 * m + 0] = l0;
        out[2 * m + 1] = l1;
        out[2 * B + 2 * m + 0] = e0 * inv;
        out[2 * B + 2 * m + 1] = e1 * inv;
    }
}

extern "C" void kernel_launch(void* const* d_in, const int* in_sizes, int n_in,
                              void* d_out, int out_size, void* d_ws, size_t ws_size,
                              hipStream_t stream) {
    (void)n_in; (void)out_size; (void)d_ws; (void)ws_size;
    const float* x  = (const float*)d_in[0];
    const float* W1 = (const float*)d_in[1];
    const float* b1 = (const float*)d_in[2];
    const float* W2 = (const float*)d_in[3];
    const float* b2 = (const float*)d_in[4];
    const float* Wf = (const float*)d_in[5];
    const float* bf = (const float*)d_in[6];
    const int*   t  = (const int*)d_in[7];
    float* out = (float*)d_out;

    int B = in_sizes[0] / 2;                  // x is [B,2]
    int grid = B / BATCH_PER_BLOCK;           // B=131072 -> 1024 blocks, all waves full
    node_rk4_wmma_kernel<<<grid, THREADS, 0, stream>>>(x, W1, b1, W2, b2, Wf, bf, t, out, B);
}